// CorrelatedGraphConv_73358041415912
// MI455X (gfx1250) — compile-verified
//
#include <hip/hip_runtime.h>
#include <hip/hip_bf16.h>

// Shapes: B=16, N=100, D=1024, L=2000
#define BB 16
#define NN 100
#define DD 1024
#define MM (BB * NN)      // 1600 rows, = 25 * 64 exactly
#define LDSK 36           // padded LDS K-stride (avoids 8-way bank conflicts)

typedef float v2f __attribute__((ext_vector_type(2)));
typedef float v8f __attribute__((ext_vector_type(8)));
typedef int   v4i __attribute__((vector_size(16)));   // matches builtin's expected pointee
typedef float4 f4;

#if __has_builtin(__builtin_amdgcn_global_load_async_to_lds_b128) && \
    __has_builtin(__builtin_amdgcn_s_wait_asynccnt)
#define USE_ASYNC_LDS 1
typedef __attribute__((address_space(1))) v4i* as1_v4i_p;
typedef __attribute__((address_space(3))) v4i* as3_v4i_p;
#else
#define USE_ASYNC_LDS 0
#endif

__device__ __forceinline__ v8f wmma_f32(v2f a, v2f b, v8f c) {
    // D = A(16x4) * B(4x16) + C, full fp32 -> v_wmma_f32_16x16x4_f32
    return __builtin_amdgcn_wmma_f32_16x16x4_f32(
        false, a, false, b, (short)0, c, false, false);
}

// 16-byte global -> LDS stage; async (no VGPR round-trip, ASYNCcnt) when available
__device__ __forceinline__ void stage16(const float* __restrict__ g, float* l) {
#if USE_ASYNC_LDS
    __builtin_amdgcn_global_load_async_to_lds_b128((as1_v4i_p)g, (as3_v4i_p)l, 0, 0);
#else
    *(f4*)l = *(const f4*)g;
#endif
}

__device__ __forceinline__ void stage_drain() {
#if USE_ASYNC_LDS
    __builtin_amdgcn_s_wait_asynccnt(0);   // all this wave's async LDS writes landed
#endif
}

// ---------------- Stage 0a: Wc = W0 + W1 + I ----------------
__global__ void prep_wc_kernel(const float* __restrict__ W0,
                               const float* __restrict__ W1,
                               float* __restrict__ Wc) {
    int t = blockIdx.x * blockDim.x + threadIdx.x;
    if (t >= DD * DD) return;
    int row = t >> 10, col = t & (DD - 1);
    Wc[t] = W0[t] + W1[t] + (row == col ? 1.0f : 0.0f);
}

// ---------------- Stage 0b: biasSum[r,:] = sum_j bias[graph[r,j],:] ----------------
__global__ __launch_bounds__(256) void bias_sum_kernel(const int* __restrict__ graph,
                                                       const float* __restrict__ bias,
                                                       float* __restrict__ biasSum) {
    __shared__ int gs[NN];
    const int r = blockIdx.x;          // 0..1599
    const int tid = threadIdx.x;       // 256 threads, 4 floats each (1024 cols)
    if (tid < NN) gs[tid] = graph[(size_t)r * NN + tid];
    __syncthreads();
    const f4* bias4 = (const f4*)bias; // row stride = 256 float4
    f4 acc = {0.f, 0.f, 0.f, 0.f};
    for (int j = 0; j < NN; ++j) {
        f4 v = bias4[(size_t)gs[j] * (DD / 4) + tid];
        acc.x += v.x; acc.y += v.y; acc.z += v.z; acc.w += v.w;
    }
    ((f4*)biasSum)[(size_t)r * (DD / 4) + tid] = acc;
}

// ---------------- Stage 1: fused 3-weight GEMM ----------------
// X[1600,1024] times Wc^T / Wa^T / Wb^T.  out[r,n] = sum_k X[r,k]*W[n,k].
// Block = 256 threads (8 waves), tile 64(M) x 64(N), K chunks of 32 in LDS.
// Wave (waveM 0..3, waveN 0..1) computes 16 rows x 32 cols (2 n-subtiles) x 3 outputs.
__global__ __launch_bounds__(256) void gemm3_kernel(
    const float* __restrict__ X,
    const float* __restrict__ Wc, const float* __restrict__ Wa, const float* __restrict__ Wb,
    const float* __restrict__ biasSum,
    const float* __restrict__ ba, const float* __restrict__ bb,
    float* __restrict__ out0, float* __restrict__ aOut, float* __restrict__ bOut)
{
    __shared__ float Xs[64 * LDSK];
    __shared__ float Ws[3][64 * LDSK];

    const int n0 = blockIdx.x * 64;
    const int m0 = blockIdx.y * 64;
    const int tid = threadIdx.x;
    const int lane = tid & 31;
    const int wave = tid >> 5;
    const int waveM = wave >> 1;      // 0..3
    const int waveN = wave & 1;       // 0..1
    const int lhi = lane >> 4;        // 0/1 (K sub-pair select)
    const int llo = lane & 15;        // M / N within tile

    v8f accC[2], accA[2], accB[2];
    #pragma unroll
    for (int s = 0; s < 2; ++s) { accC[s] = (v8f){}; accA[s] = (v8f){}; accB[s] = (v8f){}; }

    for (int k0 = 0; k0 < DD; k0 += 32) {
        __syncthreads();
        // stage X tile: 64x32 floats = 512 x b128, 2 per thread (async -> LDS)
        #pragma unroll
        for (int rep = 0; rep < 2; ++rep) {
            int lin = (tid + rep * 256) * 4;
            int row = lin >> 5, col = lin & 31;
            stage16(X + (size_t)(m0 + row) * DD + k0 + col, &Xs[row * LDSK + col]);
        }
        // stage the three weight tiles (rows = output cols n, k-contiguous)
        const float* const Wp[3] = {Wc, Wa, Wb};
        #pragma unroll
        for (int w = 0; w < 3; ++w) {
            #pragma unroll
            for (int rep = 0; rep < 2; ++rep) {
                int lin = (tid + rep * 256) * 4;
                int row = lin >> 5, col = lin & 31;
                stage16(Wp[w] + (size_t)(n0 + row) * DD + k0 + col,
                        &Ws[w][row * LDSK + col]);
            }
        }
        stage_drain();
        __syncthreads();

        const float* xw = &Xs[(waveM * 16 + llo) * LDSK + lhi * 2];
        #pragma unroll
        for (int kk = 0; kk < 8; ++kk) {              // 8 x K=4 steps = K chunk 32
            v2f af = *(const v2f*)(xw + kk * 4);
            #pragma unroll
            for (int s = 0; s < 2; ++s) {
                const int nrow = (waveN * 32 + s * 16 + llo) * LDSK + kk * 4 + lhi * 2;
                v2f bc = *(const v2f*)(&Ws[0][nrow]);
                accC[s] = wmma_f32(af, bc, accC[s]);
                v2f bA = *(const v2f*)(&Ws[1][nrow]);
                accA[s] = wmma_f32(af, bA, accA[s]);
                v2f bBv = *(const v2f*)(&Ws[2][nrow]);
                accB[s] = wmma_f32(af, bBv, accB[s]);
            }
        }
    }

    // epilogue: C/D layout -> VGPR v: row = v + lhi*8, col = llo
    #pragma unroll
    for (int s = 0; s < 2; ++s) {
        int ncol = n0 + waveN * 32 + s * 16 + llo;
        float bav = ba[ncol], bbv = bb[ncol];
        #pragma unroll
        for (int v = 0; v < 8; ++v) {
            int r = m0 + waveM * 16 + v + lhi * 8;
            size_t off = (size_t)r * DD + ncol;
            out0[off] = accC[s][v] + biasSum[off];
            aOut[off] = accA[s][v] + bav;
            bOut[off] = accB[s][v] + bbv;
        }
    }
}

// ---------------- Stage 2: alphaRaw = relu(a @ b^T) per batch ----------------
// One wave per 16x16 output tile; K=1024 in steps of 4. Rows/cols >= 100 clamped, not stored.
__global__ __launch_bounds__(32) void alpha_raw_kernel(const float* __restrict__ aBuf,
                                                       const float* __restrict__ bBuf,
                                                       float* __restrict__ alphaRaw) {
    const int b = blockIdx.z;
    const int i0 = blockIdx.y * 16, j0 = blockIdx.x * 16;
    const int lane = threadIdx.x;
    const int lhi = lane >> 4, llo = lane & 15;
    const int ri = min(i0 + llo, NN - 1);
    const int rj = min(j0 + llo, NN - 1);
    const float* ap = aBuf + ((size_t)b * NN + ri) * DD + lhi * 2;
    const float* bp = bBuf + ((size_t)b * NN + rj) * DD + lhi * 2;
    v8f acc = (v8f){};
    for (int k = 0; k < DD; k += 4) {
        v2f af = *(const v2f*)(ap + k);
        v2f bf = *(const v2f*)(bp + k);
        acc = wmma_f32(af, bf, acc);
    }
    #pragma unroll
    for (int v = 0; v < 8; ++v) {
        int i = i0 + v + lhi * 8, j = j0 + llo;
        if (i < NN && j < NN)
            alphaRaw[((size_t)b * NN + i) * NN + j] = fmaxf(acc[v], 0.0f);
    }
}

// ---------------- Stage 3: alpha2 = adj @ alphaRaw (N=100, scalar) ----------------
__global__ void adj_bmm_kernel(const int* __restrict__ graph,
                               const float* __restrict__ alphaRaw,
                               float* __restrict__ alpha2) {
    int t = blockIdx.x * blockDim.x + threadIdx.x;
    if (t >= BB * NN * NN) return;
    int k = t % NN;
    int i = (t / NN) % NN;
    int b = t / (NN * NN);
    const int* g = graph + ((size_t)b * NN + i) * NN;
    const float* ar = alphaRaw + (size_t)b * NN * NN + k;   // column k, stride NN
    float s = 0.0f;
    for (int j = 0; j < NN; ++j)
        if (g[j] != 0) s += ar[(size_t)j * NN];
    alpha2[t] = s;
}

// ---------------- Stage 4: softmax over axis=1 (over i), in place ----------------
__global__ void softmax_kernel(float* __restrict__ alpha2) {
    int t = blockIdx.x * blockDim.x + threadIdx.x;
    if (t >= BB * NN) return;
    int b = t / NN, k = t % NN;
    float* col = alpha2 + (size_t)b * NN * NN + k;          // stride-NN column; k coalesces
    float m = -3.402823466e+38f;
    for (int i = 0; i < NN; ++i) m = fmaxf(m, col[(size_t)i * NN]);
    float s = 0.0f;
    for (int i = 0; i < NN; ++i) s += __expf(col[(size_t)i * NN] - m);
    float inv = 1.0f / s;
    for (int i = 0; i < NN; ++i) col[(size_t)i * NN] = __expf(col[(size_t)i * NN] - m) * inv;
}

// ---------------- Stage 5: out = alphaSM @ out0 (K=100 = 25*4 exactly) ----------------
__global__ __launch_bounds__(32) void final_bmm_kernel(const float* __restrict__ alphaSM,
                                                       const float* __restrict__ out0,
                                                       float* __restrict__ out) {
    const int b = blockIdx.z;
    const int i0 = blockIdx.y * 16;
    const int n0 = blockIdx.x * 16;
    const int lane = threadIdx.x;
    const int lhi = lane >> 4, llo = lane & 15;
    const int ri = min(i0 + llo, NN - 1);
    const float* ap = alphaSM + (size_t)b * NN * NN + (size_t)ri * NN + lhi * 2;
    const float* bp = out0 + (size_t)b * NN * DD + n0 + llo;
    v8f acc = (v8f){};
    for (int k = 0; k < NN; k += 4) {
        v2f af = *(const v2f*)(ap + k);
        int kb = k + lhi * 2;                // B[k][n] = out0[b,k,n]; k rows not contiguous
        v2f bf;
        bf.x = bp[(size_t)kb * DD];
        bf.y = bp[(size_t)(kb + 1) * DD];
        acc = wmma_f32(af, bf, acc);
    }
    #pragma unroll
    for (int v = 0; v < 8; ++v) {
        int i = i0 + v + lhi * 8;
        if (i < NN)
            out[((size_t)b * NN + i) * DD + n0 + llo] = acc[v];
    }
}

extern "C" void kernel_launch(void* const* d_in, const int* in_sizes, int n_in,
                              void* d_out, int out_size, void* d_ws, size_t ws_size,
                              hipStream_t stream) {
    const float* feature = (const float*)d_in[0];   // [16,100,1024]
    const int*   graph   = (const int*)  d_in[1];   // [16,100,100]
    const float* W0      = (const float*)d_in[2];   // [1024,1024]
    const float* W1      = (const float*)d_in[3];
    const float* bias    = (const float*)d_in[4];   // [2000,1024]
    const float* dp_Wa   = (const float*)d_in[5];
    const float* dp_ba   = (const float*)d_in[6];   // [1024]
    const float* dp_Wb   = (const float*)d_in[7];
    const float* dp_bb   = (const float*)d_in[8];
    float* out = (float*)d_out;                     // [16,100,1024]

    // workspace layout (floats); total ~31.7 MB
    float* ws = (float*)d_ws;
    float* wc       = ws;                            // 1024*1024
    float* biasSum  = wc      + (size_t)DD * DD;     // 1600*1024
    float* aBuf     = biasSum + (size_t)MM * DD;     // 1600*1024
    float* bBuf     = aBuf    + (size_t)MM * DD;     // 1600*1024
    float* out0     = bBuf    + (size_t)MM * DD;     // 1600*1024
    float* alphaRaw = out0    + (size_t)MM * DD;     // 16*100*100
    float* alpha2   = alphaRaw + (size_t)BB * NN * NN;

    prep_wc_kernel<<<(DD * DD + 255) / 256, 256, 0, stream>>>(W0, W1, wc);
    bias_sum_kernel<<<MM, 256, 0, stream>>>(graph, bias, biasSum);
    gemm3_kernel<<<dim3(DD / 64, MM / 64), 256, 0, stream>>>(
        feature, wc, dp_Wa, dp_Wb, biasSum, dp_ba, dp_bb, out0, aBuf, bBuf);
    alpha_raw_kernel<<<dim3(7, 7, BB), 32, 0, stream>>>(aBuf, bBuf, alphaRaw);
    adj_bmm_kernel<<<(BB * NN * NN + 255) / 256, 256, 0, stream>>>(graph, alphaRaw, alpha2);
    softmax_kernel<<<(BB * NN + 255) / 256, 256, 0, stream>>>(alpha2);
    final_bmm_kernel<<<dim3(DD / 16, 7, BB), 32, 0, stream>>>(alpha2, out0, out);
}